// PersLay_19524921327763
// MI455X (gfx1250) — compile-verified
//
#include <hip/hip_runtime.h>

typedef __attribute__((ext_vector_type(2))) float v2f;
typedef __attribute__((ext_vector_type(8))) float v8f;

#define QF 64
#define CHUNK 256
#define WAVES_PER_BLOCK 8

__global__ void perslay_zero(float* __restrict__ out, int n) {
  int i = blockIdx.x * blockDim.x + threadIdx.x;
  if (i < n) out[i] = 0.0f;
}

__global__ __launch_bounds__(256)
void perslay_wmma_kernel(const float* __restrict__ input,     // [N,2]
                         const int*   __restrict__ pidx,      // [N] sorted
                         const float* __restrict__ sp,        // [2,Q]
                         const float* __restrict__ sis,       // [2,Q]
                         float* __restrict__ out,             // [D,Q]
                         int N, int D)
{
  const unsigned tid  = threadIdx.x;
  const unsigned lane = tid & 31u;
  const unsigned h    = lane >> 4;      // which half of the wave
  const unsigned m    = lane & 15u;     // A row / B,D column within 16-tile

  const int wslot = __builtin_amdgcn_readfirstlane((int)(tid >> 5));
  const long wave = (long)blockIdx.x * WAVES_PER_BLOCK + wslot;
  long pstart = wave * (long)CHUNK;
  if (pstart >= N) return;
  long pend = pstart + CHUNK; if (pend > N) pend = N;

  // Per-lane sample-point constants for this lane's 4 q-columns (q = 16t + m).
  float s0[4], s1[4], i0[4], i1[4];
#pragma unroll
  for (int t = 0; t < 4; ++t) {
    int q = t * 16 + (int)m;
    s0[t] = sp[q];        s1[t] = sp[QF + q];
    i0[t] = sis[q];       i1[t] = sis[QF + q];
  }

  v8f acc0 = {0,0,0,0,0,0,0,0};
  v8f acc1 = {0,0,0,0,0,0,0,0};
  v8f acc2 = {0,0,0,0,0,0,0,0};
  v8f acc3 = {0,0,0,0,0,0,0,0};

  // Flush 16x64 accumulator tile to out[base .. base+15][:] via f32 atomics.
  auto flush = [&](int base) {
#pragma unroll
    for (int r = 0; r < 8; ++r) {
      int row = base + r + 8 * (int)h;   // D-matrix: VGPR r holds M=r (lanes 0-15), M=r+8 (lanes 16-31)
      if (row < D) {
        float* o = out + (long)row * QF + (int)m;
        __hip_atomic_fetch_add(o +  0, acc0[r], __ATOMIC_RELAXED, __HIP_MEMORY_SCOPE_AGENT);
        __hip_atomic_fetch_add(o + 16, acc1[r], __ATOMIC_RELAXED, __HIP_MEMORY_SCOPE_AGENT);
        __hip_atomic_fetch_add(o + 32, acc2[r], __ATOMIC_RELAXED, __HIP_MEMORY_SCOPE_AGENT);
        __hip_atomic_fetch_add(o + 48, acc3[r], __ATOMIC_RELAXED, __HIP_MEMORY_SCOPE_AGENT);
      }
    }
    acc0 = (v8f){0,0,0,0,0,0,0,0};
    acc1 = (v8f){0,0,0,0,0,0,0,0};
    acc2 = (v8f){0,0,0,0,0,0,0,0};
    acc3 = (v8f){0,0,0,0,0,0,0,0};
  };

  int base = pidx[pstart];

  for (long k0 = pstart; k0 < pend; k0 += 4) {
    long kl = pend - k0;                 // valid points in group: 1..4
    bool v1 = kl > 1, v2 = kl > 2, v3 = kl > 3;

    // Uniform loads of the group's segment ids (clamped -> SMEM-friendly).
    int id0 = pidx[k0];
    int id1 = pidx[v1 ? k0 + 1 : k0];
    int id2 = pidx[v2 ? k0 + 2 : k0];
    int id3 = pidx[v3 ? k0 + 3 : k0];

    // Eager rebase so the common case finishes in one pass.
    if (id0 - base > 15) { flush(base); base = id0; }

    // This lane's two K-slots: points k0+2h and k0+2h+1 (A and B share this mapping).
    bool va = h ? v2 : true;
    bool vb = h ? v3 : v1;
    long ka = k0 + 2 * (long)h;
    long ka_s = va ? ka     : pstart;
    long kb_s = vb ? ka + 1 : pstart;
    float xa0 = input[2 * ka_s], xa1 = input[2 * ka_s + 1];
    float xb0 = input[2 * kb_s], xb1 = input[2 * kb_s + 1];

    // phi for both points across the 4 q-tiles (1 v_exp_f32 per value).
    float pa[4], pb[4];
#pragma unroll
    for (int t = 0; t < 4; ++t) {
      float za0 = (xa0 - s0[t]) * i0[t], za1 = (xa1 - s1[t]) * i1[t];
      float zb0 = (xb0 - s0[t]) * i0[t], zb1 = (xb1 - s1[t]) * i1[t];
      pa[t] = __builtin_amdgcn_exp2f(-1.442695041f * (za0 * za0 + za1 * za1));
      pb[t] = __builtin_amdgcn_exp2f(-1.442695041f * (zb0 * zb0 + zb1 * zb1));
    }
    v2f B0 = {pa[0], pb[0]};
    v2f B1 = {pa[1], pb[1]};
    v2f B2 = {pa[2], pb[2]};
    v2f B3 = {pa[3], pb[3]};

    // Bounded pass loop: handles >15-segment jumps inside a single group exactly.
    bool p0 = false, p1 = false, p2 = false, p3 = false;
#pragma unroll 1
    for (int pass = 0; pass < 4; ++pass) {
      int r0 = (!p0)       ? id0 - base : -1;
      int r1 = (v1 && !p1) ? id1 - base : -1;
      int r2 = (v2 && !p2) ? id2 - base : -1;
      int r3 = (v3 && !p3) ? id3 - base : -1;

      int rA = h ? r2 : r0;
      int rB = h ? r3 : r1;
      v2f A;
      A.x = ((unsigned)rA <= 15u && (int)m == rA) ? 1.0f : 0.0f;  // one-hot self-masks
      A.y = ((unsigned)rB <= 15u && (int)m == rB) ? 1.0f : 0.0f;

      acc0 = __builtin_amdgcn_wmma_f32_16x16x4_f32(false, A, false, B0, (short)0, acc0, false, false);
      acc1 = __builtin_amdgcn_wmma_f32_16x16x4_f32(false, A, false, B1, (short)0, acc1, false, false);
      acc2 = __builtin_amdgcn_wmma_f32_16x16x4_f32(false, A, false, B2, (short)0, acc2, false, false);
      acc3 = __builtin_amdgcn_wmma_f32_16x16x4_f32(false, A, false, B3, (short)0, acc3, false, false);

      p0 |= (unsigned)r0 <= 15u;
      p1 |= (unsigned)r1 <= 15u;
      p2 |= (unsigned)r2 <= 15u;
      p3 |= (unsigned)r3 <= 15u;

      bool rem0 = !p0, rem1 = v1 && !p1, rem2 = v2 && !p2, rem3 = v3 && !p3;
      if (!(rem0 | rem1 | rem2 | rem3)) break;
      flush(base);
      base = rem0 ? id0 : rem1 ? id1 : rem2 ? id2 : id3;  // sorted -> min remaining
    }
  }

  flush(base);
}

extern "C" void kernel_launch(void* const* d_in, const int* in_sizes, int n_in,
                              void* d_out, int out_size, void* d_ws, size_t ws_size,
                              hipStream_t stream) {
  const float* input = (const float*)d_in[0];   // [N,2] f32
  const int*   pidx  = (const int*)d_in[1];     // [N] i32 sorted
  const float* sp    = (const float*)d_in[2];   // [2,Q] f32
  const float* sis   = (const float*)d_in[3];   // [2,Q] f32
  float* out = (float*)d_out;

  int N  = in_sizes[0] / 2;
  int Qn = in_sizes[2] / 2;                     // == 64
  int D  = out_size / Qn;

  perslay_zero<<<(out_size + 255) / 256, 256, 0, stream>>>(out, out_size);

  long nwaves  = ((long)N + CHUNK - 1) / CHUNK;
  int  nblocks = (int)((nwaves + WAVES_PER_BLOCK - 1) / WAVES_PER_BLOCK);
  perslay_wmma_kernel<<<nblocks, 256, 0, stream>>>(input, pidx, sp, sis, out, N, D);
}